// Conv1dLayer_83004537963175
// MI455X (gfx1250) — compile-verified
//
#include <hip/hip_runtime.h>
#include <hip/hip_bf16.h>
#include <math.h>

// Problem constants (match reference)
#define BATCH 16
#define CIN   32
#define COUT  32
#define TLEN  65536
#define WDIM  512
#define KW    3
#define CK    (CIN * KW)        // 96 reduction length
#define ALPHA 0.2f

// Tiling
#define TT        256           // time steps per block
#define XS_BULK   4             // bulk starts at +4 floats -> 16B-aligned LDS rows
#define XS_STRIDE 264           // row stride (floats): 16B-aligned, >= TT + halo
#define WS_STRIDE 98            // >= CK, even (8B align for paired A loads)

typedef float v2f __attribute__((ext_vector_type(2)));
typedef float v8f __attribute__((ext_vector_type(8)));

__device__ __forceinline__ float lrelu(float v) {
    return v >= 0.0f ? v : ALPHA * v;
}

// ---------------------------------------------------------------------------
// Kernel 1: style affine + modulate + demodulate -> wmod[b][co][ci*3+k] in ws
// ---------------------------------------------------------------------------
__global__ __launch_bounds__(256)
void modw_kernel(const float* __restrict__ style,     // (B, WDIM)
                 const float* __restrict__ fc_weight, // (CIN, WDIM)
                 const float* __restrict__ fc_bias,   // (CIN,)
                 const float* __restrict__ weight,    // (COUT, CIN, K)
                 float* __restrict__ wmod)            // (B, COUT, CK)
{
    const int b   = blockIdx.x;
    const int tid = threadIdx.x;
    __shared__ float s_sh[CIN];
    __shared__ float d_sh[COUT];

    const float gain = 0.04419417382415922f; // 1/sqrt(512)

    if (tid < CIN) {
        const float* st = style + (size_t)b * WDIM;
        const float* fw = fc_weight + (size_t)tid * WDIM;
        float acc = fc_bias[tid];
        for (int i = 0; i < WDIM; ++i) acc += st[i] * (fw[i] * gain);
        s_sh[tid] = lrelu(acc);
    }
    __syncthreads();

    if (tid < COUT) {
        const float* wr = weight + (size_t)tid * CK;
        float sq = 0.0f;
        for (int c = 0; c < CK; ++c) {
            float v = wr[c] * s_sh[c / KW];
            sq += v * v;
        }
        d_sh[tid] = rsqrtf(sq + 1e-8f);
    }
    __syncthreads();

    float* wout = wmod + (size_t)b * COUT * CK;
    for (int idx = tid; idx < COUT * CK; idx += 256) {
        int co = idx / CK;
        int c  = idx - co * CK;
        wout[idx] = weight[idx] * s_sh[c / KW] * d_sh[co];
    }
}

// ---------------------------------------------------------------------------
// Kernel 2: grouped conv1d as GEMM via V_WMMA_F32_16X16X4_F32
//   Y[b][co][t] = sum_{c in 0..95} Wmod[b][co][c] * x[b][c/3][t + c%3 - 1]
// Block: (batch b, 256-time tile). 8 waves; wave w owns times [w*32, w*32+32).
// x tile staged via GLOBAL_LOAD_ASYNC_TO_LDS_B128 (ASYNCcnt), halo via scalar.
// Each wave: 2 co-tiles x 2 t-tiles accumulators, 24 K-steps -> 96 WMMA.
// ---------------------------------------------------------------------------
__global__ __launch_bounds__(256)
void conv_wmma_kernel(const float* __restrict__ x,      // (B, CIN, T)
                      const float* __restrict__ wmod,   // (B, COUT, CK)
                      const float* __restrict__ bias,   // (COUT,)
                      const float* __restrict__ nstr,   // (COUT,)
                      const float* __restrict__ noise,  // (B, 1, T)
                      float* __restrict__ out)          // (B, COUT, T)
{
    __shared__ float xs[CIN * XS_STRIDE];      // x tile + halo, 16B-aligned rows
    __shared__ float wsh[COUT * WS_STRIDE];    // modulated weights
    __shared__ float bias_sh[COUT];
    __shared__ float nstr_sh[COUT];

    const int b   = blockIdx.y;
    const int t0  = blockIdx.x * TT;
    const int tid = threadIdx.x;

    // ---- stage modulated weights + per-channel epilogue vectors ----
    const float* wb = wmod + (size_t)b * COUT * CK;
    for (int idx = tid; idx < COUT * CK; idx += 256) {
        int co = idx / CK;
        int c  = idx - co * CK;
        wsh[co * WS_STRIDE + c] = wb[idx];
    }
    if (tid < COUT) {
        bias_sh[tid] = bias[tid];
        nstr_sh[tid] = nstr[tid];
    }

    // ---- stage x bulk [t0, t0+TT) via async global->LDS 16B copies ----
    // 32 rows x 64 b128-transfers = 2048 transfers, 8 per thread. Always in
    // range (t0 multiple of TT). Both global and LDS addresses 16B-aligned.
    const float* xb = x + (size_t)b * CIN * TLEN;
#pragma unroll
    for (int it = 0; it < (CIN * (TT / 4)) / 256; ++it) {   // 8 iterations
        const int idx = tid + it * 256;
        const int ci  = idx >> 6;          // 64 x b128 per channel row
        const int j   = idx & 63;
        const float* gp = xb + (size_t)ci * TLEN + t0 + j * 4;
        const unsigned lds_off =
            (unsigned)(uintptr_t)&xs[ci * XS_STRIDE + XS_BULK + j * 4];
        asm volatile("global_load_async_to_lds_b128 %0, %1, off"
                     :: "v"(lds_off), "v"(gp)
                     : "memory");
    }
    // ---- halo columns (t0-1 and t0+TT), zero-padded at tensor edges ----
    if (tid < 2 * CIN) {
        const int ci   = tid >> 1;
        const int side = tid & 1;
        const int gt   = side ? (t0 + TT) : (t0 - 1);
        float v = (gt >= 0 && gt < TLEN) ? xb[(size_t)ci * TLEN + gt] : 0.0f;
        xs[ci * XS_STRIDE + (side ? (XS_BULK + TT) : (XS_BULK - 1))] = v;
    }
    asm volatile("s_wait_asynccnt 0x0" ::: "memory");
    __syncthreads();

    const int wave = tid >> 5;
    const int lane = tid & 31;
    const int tl   = wave * 32;        // local time base (two 16-wide N tiles)
    const int mn   = lane & 15;        // row (A) / column (B/D) index
    const int kb   = (lane >> 4) * 2;  // K sub-pair: lanes 0-15 -> {0,1}, 16-31 -> {2,3}

    v8f acc00 = {}; // co 0-15,  t tl+0..15
    v8f acc01 = {}; // co 0-15,  t tl+16..31
    v8f acc10 = {}; // co 16-31, t tl+0..15
    v8f acc11 = {}; // co 16-31, t tl+16..31

    const float* wrow0 = &wsh[mn * WS_STRIDE];
    const float* wrow1 = &wsh[(mn + 16) * WS_STRIDE];
    const int tbase = tl + mn + (XS_BULK - 1);   // halo coord base (k=0 -> t-1)

#pragma unroll
    for (int s = 0; s < CK / 4; ++s) {
        const int c0 = 4 * s + kb;
        const int c1 = c0 + 1;

        // A fragments (ISA 16x4 f32 layout): consecutive -> paired load
        v2f a0, a1;
        a0.x = wrow0[c0]; a0.y = wrow0[c1];
        a1.x = wrow1[c0]; a1.y = wrow1[c1];

        // B fragments: im2col rows c0,c1 -> xs[ci][tbase + koff]
        const int ci0 = c0 / KW, k0 = c0 - ci0 * KW;
        const int ci1 = c1 / KW, k1 = c1 - ci1 * KW;
        const int o0 = ci0 * XS_STRIDE + tbase + k0;
        const int o1 = ci1 * XS_STRIDE + tbase + k1;
        v2f b0, b1;
        b0.x = xs[o0];      b0.y = xs[o1];
        b1.x = xs[o0 + 16]; b1.y = xs[o1 + 16];

        acc00 = __builtin_amdgcn_wmma_f32_16x16x4_f32(false, a0, false, b0,
                                                      (short)0, acc00, false, false);
        acc01 = __builtin_amdgcn_wmma_f32_16x16x4_f32(false, a0, false, b1,
                                                      (short)0, acc01, false, false);
        acc10 = __builtin_amdgcn_wmma_f32_16x16x4_f32(false, a1, false, b0,
                                                      (short)0, acc10, false, false);
        acc11 = __builtin_amdgcn_wmma_f32_16x16x4_f32(false, a1, false, b1,
                                                      (short)0, acc11, false, false);
    }

    // ---- epilogue: noise*strength + bias, leaky relu, store ----
    // D layout: VGPR r, lanes 0-15 -> (M=r, N=lane); lanes 16-31 -> (M=r+8, N=lane-16)
    const int    halfofs = (lane >> 4) * 8;
    const float* nz = noise + (size_t)b * TLEN;
    float* ob = out + (size_t)b * COUT * TLEN;

    const int ta  = t0 + tl + mn;       // t for *-tile 0
    const int tbb = ta + 16;            // t for *-tile 1
    const float noi0 = nz[ta];
    const float noi1 = nz[tbb];

#pragma unroll
    for (int vr = 0; vr < 8; ++vr) {
        const int coA = vr + halfofs;       // co tile 0
        const int coB = coA + 16;           // co tile 1
        const float bA = bias_sh[coA], sA = nstr_sh[coA];
        const float bB = bias_sh[coB], sB = nstr_sh[coB];

        ob[(size_t)coA * TLEN + ta]  = lrelu(acc00[vr] + noi0 * sA + bA);
        ob[(size_t)coA * TLEN + tbb] = lrelu(acc01[vr] + noi1 * sA + bA);
        ob[(size_t)coB * TLEN + ta]  = lrelu(acc10[vr] + noi0 * sB + bB);
        ob[(size_t)coB * TLEN + tbb] = lrelu(acc11[vr] + noi1 * sB + bB);
    }
}

// ---------------------------------------------------------------------------
// Launch
// ---------------------------------------------------------------------------
extern "C" void kernel_launch(void* const* d_in, const int* in_sizes, int n_in,
                              void* d_out, int out_size, void* d_ws, size_t ws_size,
                              hipStream_t stream) {
    const float* x          = (const float*)d_in[0];
    const float* style      = (const float*)d_in[1];
    const float* fc_weight  = (const float*)d_in[2];
    const float* fc_bias    = (const float*)d_in[3];
    const float* weight     = (const float*)d_in[4];
    const float* bias       = (const float*)d_in[5];
    const float* nstr       = (const float*)d_in[6];
    const float* noise      = (const float*)d_in[7];
    float* out  = (float*)d_out;
    float* wmod = (float*)d_ws;   // B*COUT*CK floats = 196 KB scratch

    modw_kernel<<<BATCH, 256, 0, stream>>>(style, fc_weight, fc_bias, weight, wmod);

    dim3 grid(TLEN / TT, BATCH, 1);
    conv_wmma_kernel<<<grid, 256, 0, stream>>>(x, wmod, bias, nstr, noise, out);
}